// Efficient_Linear_Embedding_Dimensional_Attention_53824530153954
// MI455X (gfx1250) — compile-verified
//
#include <hip/hip_runtime.h>
#include <math.h>

typedef __attribute__((ext_vector_type(2))) float v2f;
typedef __attribute__((ext_vector_type(8))) float v8f;

#define WMMA_F32(a, b, c) \
  __builtin_amdgcn_wmma_f32_16x16x4_f32(false, (a), false, (b), (short)0, (c), false, false)

static constexpr int Bsz    = 64;
static constexpr int Npatch = 1024;
static constexpr int Cch    = 768;
static constexpr int NH     = 8;
static constexpr int HD     = 128;   // Npatch / NH

// ---------------------------------------------------------------------------
// GEMM: Out[b] = W @ X[b]  (W row-major [1024,1024], X[b] row-major [1024,768])
// Optional bias per output row (proj) and per-block sum-of-squares partials (K).
// Block = 256 threads = 8 waves; tile 128x128; each wave owns 32x64 (2x4 WMMA
// accumulators). Double-buffered LDS staging, one barrier per 16-wide K tile.
// B is staged K-paired: Bs[p][c] = {B[2p][c], B[2p+1][c]} so every WMMA B
// fragment is a single contiguous ds_load_b64 (no repacking movs).
// ---------------------------------------------------------------------------
__global__ __launch_bounds__(256) void gemm_wx_kernel(
    const float* __restrict__ W, const float* __restrict__ X,
    float* __restrict__ Out, const float* __restrict__ bias,
    float* __restrict__ partials)
{
  constexpr int LDA = 18;        // 16 + 2 pad: conflict-free A-fragment reads
  constexpr int ASZ = 128 * LDA; // 2304 floats per buffer
  constexpr int PB  = 288;       // paired-B pitch: 288 % 64 == 32 -> halves on
                                 // disjoint bank halves for fragment b64 reads
  constexpr int BSZ = 8 * PB;    // 2304 floats per buffer
  __shared__ __align__(16) float As[2 * ASZ];
  __shared__ __align__(16) float Bs[2 * BSZ];
  __shared__ float red[256];

  const int t    = threadIdx.x;
  const int lane = t & 31;
  const int wave = t >> 5;
  const int wr   = wave >> 1;          // 0..3 : 32-row group
  const int wc   = wave & 1;           // 0..1 : 64-col group
  const int lm   = lane & 15;          // fragment row/col index
  const int lk   = (lane >> 4) << 1;   // K offset 0 / 2 (A,B frag layout)
  const int lmh  = (lane >> 4) << 3;   // C/D row offset 0 / 8

  const int col0 = blockIdx.x * 128;
  const int row0 = blockIdx.y * 128;
  const int b    = blockIdx.z;

  const float* Xb = X + (size_t)b * Npatch * Cch;
  float*       Ob = Out + (size_t)b * Npatch * Cch;

  v8f acc[2][4] = {};

  // staging assignments
  const int ra  = t >> 1;        // 0..127 : A row
  const int ca  = (t & 1) * 8;   // 0/8    : A col group
  const int pbi = t >> 5;        // 0..7   : B K-pair index
  const int cbb = (t & 31) * 4;  // 0..124 : B col group (float4)

  const float* wA0 = W + (size_t)(row0 + ra) * Npatch + ca;
  const float* xB0 = Xb + (size_t)(2 * pbi) * Cch + col0 + cbb;

  // prologue: stage first K tile into buffer 0
  float4 a0 = *(const float4*)(wA0);
  float4 a1 = *(const float4*)(wA0 + 4);
  float4 b0 = *(const float4*)(xB0);
  float4 b1 = *(const float4*)(xB0 + Cch);
  {
    float* ap = As + ra * LDA + ca;
    ap[0] = a0.x; ap[1] = a0.y; ap[2] = a0.z; ap[3] = a0.w;
    ap[4] = a1.x; ap[5] = a1.y; ap[6] = a1.z; ap[7] = a1.w;
    float* bp = Bs + pbi * PB + 2 * cbb;
    *(float4*)(bp)     = make_float4(b0.x, b1.x, b0.y, b1.y);
    *(float4*)(bp + 4) = make_float4(b0.z, b1.z, b0.w, b1.w);
  }
  __syncthreads();

  int buf = 0;
  for (int k0 = 0; k0 < Npatch; k0 += 16) {
    const bool more = (k0 + 16) < Npatch;
    if (more) {  // issue next tile's global loads before compute (overlap)
      a0 = *(const float4*)(wA0 + k0 + 16);
      a1 = *(const float4*)(wA0 + k0 + 20);
      b0 = *(const float4*)(xB0 + (size_t)(k0 + 16) * Cch);
      b1 = *(const float4*)(xB0 + (size_t)(k0 + 17) * Cch);
    }
    const float* Ab = As + buf * ASZ;
    const float* Bb = Bs + buf * BSZ;
#pragma unroll
    for (int kk = 0; kk < 16; kk += 4) {
      v2f af[2], bf[4];
#pragma unroll
      for (int ti = 0; ti < 2; ++ti)
        af[ti] = *(const v2f*)(Ab + (wr * 32 + ti * 16 + lm) * LDA + kk + lk);
#pragma unroll
      for (int tj = 0; tj < 4; ++tj)
        bf[tj] = *(const v2f*)(Bb + ((kk + lk) >> 1) * PB +
                               2 * (wc * 64 + tj * 16 + lm));
#pragma unroll
      for (int ti = 0; ti < 2; ++ti)
#pragma unroll
        for (int tj = 0; tj < 4; ++tj)
          acc[ti][tj] = WMMA_F32(af[ti], bf[tj], acc[ti][tj]);
    }
    if (more) {  // stage next tile into the other buffer
      float* ap = As + (buf ^ 1) * ASZ + ra * LDA + ca;
      ap[0] = a0.x; ap[1] = a0.y; ap[2] = a0.z; ap[3] = a0.w;
      ap[4] = a1.x; ap[5] = a1.y; ap[6] = a1.z; ap[7] = a1.w;
      float* bp = Bs + (buf ^ 1) * BSZ + pbi * PB + 2 * cbb;
      *(float4*)(bp)     = make_float4(b0.x, b1.x, b0.y, b1.y);
      *(float4*)(bp + 4) = make_float4(b0.z, b1.z, b0.w, b1.w);
    }
    __syncthreads();
    buf ^= 1;
  }

  float ssq = 0.0f;
#pragma unroll
  for (int ti = 0; ti < 2; ++ti)
#pragma unroll
    for (int tj = 0; tj < 4; ++tj)
#pragma unroll
      for (int r = 0; r < 8; ++r) {
        const int mrow = row0 + wr * 32 + ti * 16 + r + lmh;
        const int ncol = col0 + wc * 64 + tj * 16 + lm;
        float v = acc[ti][tj][r];
        ssq += v * v;
        if (bias) v += bias[mrow];
        Ob[(size_t)mrow * Cch + ncol] = v;
      }

  if (partials) {  // uniform branch: fixed-order deterministic block reduce
    red[t] = ssq;
    __syncthreads();
#pragma unroll
    for (int off = 128; off > 0; off >>= 1) {
      if (t < off) red[t] += red[t + off];
      __syncthreads();
    }
    if (t == 0)
      partials[(blockIdx.z * gridDim.y + blockIdx.y) * gridDim.x + blockIdx.x] = red[0];
  }
}

// ---------------------------------------------------------------------------
// Deterministic partial-sum reduction -> inv = 1/(||K||_F + EPS)
// ---------------------------------------------------------------------------
__global__ __launch_bounds__(256) void reduce_inv_kernel(
    const float* __restrict__ partials, int n, float* __restrict__ invOut)
{
  __shared__ float red[256];
  float s = 0.0f;
  for (int i = threadIdx.x; i < n; i += 256) s += partials[i];
  red[threadIdx.x] = s;
  __syncthreads();
#pragma unroll
  for (int off = 128; off > 0; off >>= 1) {
    if (threadIdx.x < off) red[threadIdx.x] += red[threadIdx.x + off];
    __syncthreads();
  }
  if (threadIdx.x == 0) invOut[0] = 1.0f / (sqrtf(red[0]) + 1e-8f);
}

// ---------------------------------------------------------------------------
// Attention per (h, b):  kt = inv^2*scale[h] * K_h @ V_h^T   [128x128]
// then Out_h = kt^T @ X_h  ([128x768]), exact GELU, write to G (aliases K:
// this workgroup is the sole reader/writer of its 128-row slice, and all K/V
// reads complete before the post-kt barrier).
// Phase 1 uses double-buffered staging with one barrier per 16-wide c tile.
// ---------------------------------------------------------------------------
__global__ __launch_bounds__(256) void attn_kernel(
    const float* __restrict__ K, const float* __restrict__ V,
    const float* __restrict__ X, const float* __restrict__ scale,
    const float* __restrict__ invp, float* __restrict__ G)
{
  constexpr int LDK = 18;        // 16 + 2 pad
  constexpr int KSZ = 128 * LDK; // 2304 floats per buffer
  constexpr int LDT = 140;       // 2*LDT%64=24, 8*LDT%64=32 -> no bank overlap
  __shared__ __align__(16) float kts[128 * LDT];
  __shared__ __align__(16) float Ks[2 * KSZ];
  __shared__ __align__(16) float Vs[2 * KSZ];

  const int h    = blockIdx.x;
  const int b    = blockIdx.y;
  const int t    = threadIdx.x;
  const int lane = t & 31;
  const int wave = t >> 5;
  const int lm   = lane & 15;
  const int lk   = (lane >> 4) << 1;
  const int lmh  = (lane >> 4) << 3;

  const size_t base = ((size_t)b * Npatch + h * HD) * Cch;
  const float* Kh = K + base;
  const float* Vh = V + base;
  const float* Xh = X + base;
  float*       Gh = G + base;

  const float inv = invp[0];
  const float f   = inv * inv * scale[h];

  v8f acc[8] = {};

  const int ra = t >> 1;
  const int ca = (t & 1) * 8;
  const float* kG = Kh + (size_t)ra * Cch + ca;
  const float* vG = Vh + (size_t)ra * Cch + ca;

  // prologue: stage c tile 0 into buffer 0
  float4 k0v = *(const float4*)(kG);
  float4 k1v = *(const float4*)(kG + 4);
  float4 v0v = *(const float4*)(vG);
  float4 v1v = *(const float4*)(vG + 4);
  {
    float* kp = Ks + ra * LDK + ca;
    kp[0] = k0v.x; kp[1] = k0v.y; kp[2] = k0v.z; kp[3] = k0v.w;
    kp[4] = k1v.x; kp[5] = k1v.y; kp[6] = k1v.z; kp[7] = k1v.w;
    float* vp = Vs + ra * LDK + ca;
    vp[0] = v0v.x; vp[1] = v0v.y; vp[2] = v0v.z; vp[3] = v0v.w;
    vp[4] = v1v.x; vp[5] = v1v.y; vp[6] = v1v.z; vp[7] = v1v.w;
  }
  __syncthreads();

  // Phase 1: kt[n,m] = sum_c K_h[n,c] * V_h[m,c]
  int buf = 0;
  for (int c0 = 0; c0 < Cch; c0 += 16) {
    const bool more = (c0 + 16) < Cch;
    if (more) {
      k0v = *(const float4*)(kG + c0 + 16);
      k1v = *(const float4*)(kG + c0 + 20);
      v0v = *(const float4*)(vG + c0 + 16);
      v1v = *(const float4*)(vG + c0 + 20);
    }
    const float* Kb = Ks + buf * KSZ;
    const float* Vb = Vs + buf * KSZ;
#pragma unroll
    for (int kk = 0; kk < 16; kk += 4) {
      // B fragment: B[c,m] = V_h[m,c] (transpose via strided per-lane read)
      const v2f bf = *(const v2f*)(Vb + (wave * 16 + lm) * LDK + kk + lk);
#pragma unroll
      for (int ti = 0; ti < 8; ++ti) {
        const v2f af = *(const v2f*)(Kb + (ti * 16 + lm) * LDK + kk + lk);
        acc[ti] = WMMA_F32(af, bf, acc[ti]);
      }
    }
    if (more) {
      float* kp = Ks + (buf ^ 1) * KSZ + ra * LDK + ca;
      kp[0] = k0v.x; kp[1] = k0v.y; kp[2] = k0v.z; kp[3] = k0v.w;
      kp[4] = k1v.x; kp[5] = k1v.y; kp[6] = k1v.z; kp[7] = k1v.w;
      float* vp = Vs + (buf ^ 1) * KSZ + ra * LDK + ca;
      vp[0] = v0v.x; vp[1] = v0v.y; vp[2] = v0v.z; vp[3] = v0v.w;
      vp[4] = v1v.x; vp[5] = v1v.y; vp[6] = v1v.z; vp[7] = v1v.w;
    }
    __syncthreads();
    buf ^= 1;
  }

#pragma unroll
  for (int ti = 0; ti < 8; ++ti)
#pragma unroll
    for (int r = 0; r < 8; ++r)
      kts[(ti * 16 + r + lmh) * LDT + wave * 16 + lm] = f * acc[ti][r];
  __syncthreads();

  // Phase 2: Out[m,c] = sum_n kt[n,m] * X_h[n,c]; wave owns m in [16w,16w+16)
  for (int cblk = 0; cblk < Cch; cblk += 64) {
    v8f o[4] = {};
    for (int nn = 0; nn < 128; nn += 4) {
      v2f af;
      af.x = kts[(nn + lk) * LDT + wave * 16 + lm];
      af.y = kts[(nn + lk + 1) * LDT + wave * 16 + lm];
#pragma unroll
      for (int tj = 0; tj < 4; ++tj) {
        const int c = cblk + tj * 16 + lm;
        v2f bf;
        bf.x = Xh[(size_t)(nn + lk) * Cch + c];
        bf.y = Xh[(size_t)(nn + lk + 1) * Cch + c];
        o[tj] = WMMA_F32(af, bf, o[tj]);
      }
    }
#pragma unroll
    for (int tj = 0; tj < 4; ++tj)
#pragma unroll
      for (int r = 0; r < 8; ++r) {
        const int m = wave * 16 + r + lmh;
        const int c = cblk + tj * 16 + lm;
        const float v = o[tj][r];
        // exact GELU: 0.5*v*(1+erf(v/sqrt(2)))
        Gh[(size_t)m * Cch + c] = 0.5f * v * (1.0f + erff(v * 0.70710678118654752440f));
      }
  }
}

// ---------------------------------------------------------------------------
extern "C" void kernel_launch(void* const* d_in, const int* in_sizes, int n_in,
                              void* d_out, int out_size, void* d_ws, size_t ws_size,
                              hipStream_t stream) {
  (void)in_sizes; (void)n_in; (void)out_size; (void)ws_size;
  const float* x      = (const float*)d_in[0];
  const float* qk_w   = (const float*)d_in[1];
  const float* v_w    = (const float*)d_in[2];
  const float* proj_w = (const float*)d_in[3];
  const float* proj_b = (const float*)d_in[4];
  const float* scale  = (const float*)d_in[5];
  float* out = (float*)d_out;

  const size_t elems = (size_t)Bsz * Npatch * Cch;  // 50,331,648
  float* Kbuf     = (float*)d_ws;                   // 201.3 MB; later reused as G
  float* partials = Kbuf + elems;                   // 3072 floats
  float* invp     = partials + 3072;                // 1 float

  const int nPart = (Cch / 128) * (Npatch / 128) * Bsz;  // 6*8*64 = 3072
  dim3 ggrid(Cch / 128, Npatch / 128, Bsz);

  // K = qk_w @ X (+ ssq partials); V = v_w @ X staged in d_out
  gemm_wx_kernel<<<ggrid, 256, 0, stream>>>(qk_w, x, Kbuf, nullptr, partials);
  gemm_wx_kernel<<<ggrid, 256, 0, stream>>>(v_w,  x, out,  nullptr, nullptr);
  // inv = 1/(||K||_F + eps), deterministic order
  reduce_inv_kernel<<<1, 256, 0, stream>>>(partials, nPart, invp);
  // kt + out + GELU, G written in-place over Kbuf
  attn_kernel<<<dim3(NH, Bsz), 256, 0, stream>>>(Kbuf, out, x, scale, invp, Kbuf);
  // final: out = proj_w @ G + proj_b  (overwrites the V staging)
  gemm_wx_kernel<<<ggrid, 256, 0, stream>>>(proj_w, Kbuf, out, proj_b, nullptr);
}